// GATLayer_33019708572037
// MI455X (gfx1250) — compile-verified
//
#include <hip/hip_runtime.h>

typedef float v2f __attribute__((ext_vector_type(2)));
typedef float v8f __attribute__((ext_vector_type(8)));
typedef int   gv4i __attribute__((vector_size(16)));   // matches builtin param type

#define N_NODES 50000
#define N_EDGES 1600000
#define IN_F    128
#define OUT_F   64
#define HEADS   2
#define COLS    (HEADS * OUT_F)   // 128
#define LRELU_ALPHA 0.2f

#define XS_STRIDE 132   // 128 + 4: lane m -> bank 4m+const, conflict-free b64 A loads
#define WS_STRIDE 72    // 64 + 8: khalf groups land 16 banks apart, conflict-free B loads

// Async global->LDS staging (CDNA5 GLOBAL_LOAD_ASYNC_TO_LDS_B128, ASYNCcnt).
// Device-pass probe confirmed: 4 args, param0 = v4i* in global AS (non-const).
// Host pass: __has_builtin false for amdgcn builtins -> fallback (never executed).
#if __has_builtin(__builtin_amdgcn_global_load_async_to_lds_b128)
#define ASYNC_CP(gsrc, ldst)                                                   \
  __builtin_amdgcn_global_load_async_to_lds_b128(                              \
      (__attribute__((address_space(1))) gv4i*)(gsrc),                         \
      (__attribute__((address_space(3))) gv4i*)(ldst), 0, 0)
#if __has_builtin(__builtin_amdgcn_s_wait_asynccnt)
#define ASYNC_WAIT() __builtin_amdgcn_s_wait_asynccnt(0)
#else
#define ASYNC_WAIT() asm volatile("s_wait_asynccnt 0" ::: "memory")
#endif
#else
#define ASYNC_CP(gsrc, ldst) (*(float4*)(ldst) = *(const float4*)(gsrc))
#define ASYNC_WAIT() do { } while (0)
#endif

// ---------------------------------------------------------------------------
// K1: Wh = x @ B, B[i][c] = W[c>>6][i][c&63].  Wh written to d_out (identical
// layout to the final output: out[n*128 + h*64 + o]).
// Block = 128 threads (4 waves); 64 rows x 128 cols per block.
// Each wave: one 16-row strip, 8 col-tiles of 16, K=128 in steps of 4 via
// V_WMMA_F32_16X16X4_F32 (exact fp32 — problem is bandwidth-bound, no reason
// to drop precision).
// A-frag (16x4 f32): lane L -> row M = L%16; VGPR j -> K = j + 2*(L/16).
// C/D: VGPR r -> row M = r + 8*(L/16); col N = L%16.
// Both operand tiles staged in padded LDS (bank-conflict-free fragment reads)
// via async global->LDS b128 copies.
// ---------------------------------------------------------------------------
__global__ __launch_bounds__(128)
void gat_gemm_wmma(const float* __restrict__ x, const float* __restrict__ W,
                   float* __restrict__ Wh) {
  __shared__ float xs[64 * XS_STRIDE];            // ~33.8 KB x-tile (padded)
  __shared__ float wsh[HEADS * IN_F * WS_STRIDE]; // ~73.7 KB W (padded rows of 64)

  const int tid    = threadIdx.x;        // 0..127
  const int lane   = tid & 31;
  const int wave   = tid >> 5;           // 0..3
  const int rowBlk = blockIdx.x * 64;

  // Stage x tile: 64 rows x 128 cols, one b128 per thread-iteration, async.
  for (int i = tid; i < 64 * 32; i += 128) {
    const int r  = i >> 5;               // local row 0..63
    const int c4 = (i & 31) * 4;         // col 0..124 step 4
    int gr = rowBlk + r;
    if (gr >= N_NODES) gr = N_NODES - 1; // clamped garbage, never stored
    ASYNC_CP(x + (size_t)gr * IN_F + c4, &xs[r * XS_STRIDE + c4]);
  }
  // Stage W: 2*128 rows of 64 floats -> padded LDS rows. 4096 b128 total.
  for (int i = tid; i < 4096; i += 128) {
    const int row = i >> 4;              // head*128 + k, 0..255
    const int o   = (i & 15) * 4;        // 0..60 step 4
    ASYNC_CP(W + row * OUT_F + o, &wsh[row * WS_STRIDE + o]);
  }
  ASYNC_WAIT();
  __syncthreads();

  const int m      = lane & 15;          // M within A tile / N within C tile
  const int khalf  = lane >> 4;          // 0 or 1
  const int rowLoc = wave * 16;

  v8f acc[8] = {};                       // 8 col-tiles x 8 VGPRs

  #pragma unroll 4
  for (int k0 = 0; k0 < IN_F; k0 += 4) {
    const int kk = k0 + 2 * khalf;
    // A fragment: 2 consecutive K -> one ds_load_b64, conflict-free (stride 132)
    const float2 av = *(const float2*)&xs[(rowLoc + m) * XS_STRIDE + kk];
    v2f a; a.x = av.x; a.y = av.y;
    #pragma unroll
    for (int t = 0; t < 8; ++t) {
      const int head   = t >> 2;         // 16-col tiles never straddle a head
      const int obase  = (t & 3) * 16 + m;
      const int wrow   = head * IN_F + kk;
      v2f b;
      b.x = wsh[(wrow + 0) * WS_STRIDE + obase];
      b.y = wsh[(wrow + 1) * WS_STRIDE + obase];
      acc[t] = __builtin_amdgcn_wmma_f32_16x16x4_f32(
          false, a, false, b, (short)0, acc[t], false, false);
    }
  }

  // 50000 % 16 == 0 -> each 16-row strip is entirely valid or entirely not:
  // single wave-uniform guard, unguarded coalesced b32 stores inside.
  if (rowBlk + rowLoc < N_NODES) {
    #pragma unroll
    for (int t = 0; t < 8; ++t) {
      const int col = t * 16 + m;
      #pragma unroll
      for (int r = 0; r < 8; ++r) {
        const int grow = rowBlk + rowLoc + r + 8 * khalf;
        Wh[(size_t)grow * COLS + col] = acc[t][r];
      }
    }
  }
}

// ---------------------------------------------------------------------------
// K2: s_i[n,h] = Wh[n,h,:]·a[h,0:64], s_j[n,h] = Wh[n,h,:]·a[h,64:128].
// Also zero-init coeff table and Z sums (d_ws keeps state between replays,
// so every call must re-zero).
// ---------------------------------------------------------------------------
__global__ __launch_bounds__(256)
void gat_scores(const float* __restrict__ Wh, const float* __restrict__ a,
                float* __restrict__ sI, float* __restrict__ sJ,
                float* __restrict__ coeff, float* __restrict__ Zsum) {
  const int t = blockIdx.x * 256 + threadIdx.x;    // t = n*2 + h
  if (t < HEADS) Zsum[t] = 0.0f;
  if (t >= N_NODES * HEADS) return;
  coeff[t] = 0.0f;
  const int n = t >> 1, h = t & 1;
  const float4* wp = (const float4*)(Wh + (size_t)n * COLS + h * OUT_F);
  const float4* ai = (const float4*)(a + h * (2 * OUT_F));
  const float4* aj = ai + (OUT_F / 4);
  float si = 0.f, sj = 0.f;
  #pragma unroll
  for (int q = 0; q < OUT_F / 4; ++q) {
    const float4 w = wp[q];
    const float4 vi = ai[q];
    const float4 vj = aj[q];
    si += w.x * vi.x + w.y * vi.y + w.z * vi.z + w.w * vi.w;
    sj += w.x * vj.x + w.y * vj.y + w.z * vj.z + w.w * vj.w;
  }
  sI[t] = si;
  sJ[t] = sj;
}

// ---------------------------------------------------------------------------
// K3: per-edge p = exp(leakyrelu(s_i[src] + s_j[dst])) for both heads.
// coeff[dst,h] += p (atomic; ~32 edges/node average contention);
// Z[h] += p via LDS tree-reduce + one atomic per block per head.
// ---------------------------------------------------------------------------
__global__ __launch_bounds__(256)
void gat_edges(const int* __restrict__ srcIdx, const int* __restrict__ dstIdx,
               const float* __restrict__ sI, const float* __restrict__ sJ,
               float* __restrict__ coeff, float* __restrict__ Zsum) {
  __shared__ float red0[256];
  __shared__ float red1[256];
  const int e = blockIdx.x * 256 + threadIdx.x;
  float p0 = 0.f, p1 = 0.f;
  if (e < N_EDGES) {
    const int s = srcIdx[e];
    const int d = dstIdx[e];
    float e0 = sI[s * 2 + 0] + sJ[d * 2 + 0];
    float e1 = sI[s * 2 + 1] + sJ[d * 2 + 1];
    e0 = e0 > 0.f ? e0 : LRELU_ALPHA * e0;
    e1 = e1 > 0.f ? e1 : LRELU_ALPHA * e1;
    p0 = __expf(e0);
    p1 = __expf(e1);
    atomicAdd(&coeff[d * 2 + 0], p0);
    atomicAdd(&coeff[d * 2 + 1], p1);
  }
  red0[threadIdx.x] = p0;
  red1[threadIdx.x] = p1;
  __syncthreads();
  for (int s2 = 128; s2 > 0; s2 >>= 1) {
    if (threadIdx.x < s2) {
      red0[threadIdx.x] += red0[threadIdx.x + s2];
      red1[threadIdx.x] += red1[threadIdx.x + s2];
    }
    __syncthreads();
  }
  if (threadIdx.x == 0) {
    atomicAdd(&Zsum[0], red0[0]);
    atomicAdd(&Zsum[1], red1[0]);
  }
}

// ---------------------------------------------------------------------------
// K4: out[n, h*64+o] *= coeff[n,h] / Z[h]   (in place, float4 per thread)
// ---------------------------------------------------------------------------
__global__ __launch_bounds__(256)
void gat_scale(float* __restrict__ out, const float* __restrict__ coeff,
               const float* __restrict__ Zsum) {
  const int i = blockIdx.x * 256 + threadIdx.x;  // float4 index
  if (i >= N_NODES * (COLS / 4)) return;
  const int n    = i >> 5;                       // 32 float4 per row
  const int head = (i >> 4) & 1;                 // 16 float4 per head
  const float zc = coeff[n * 2 + head] / Zsum[head];
  float4* p = (float4*)out + i;
  float4 v  = *p;
  v.x *= zc; v.y *= zc; v.z *= zc; v.w *= zc;
  *p = v;
}

extern "C" void kernel_launch(void* const* d_in, const int* in_sizes, int n_in,
                              void* d_out, int out_size, void* d_ws, size_t ws_size,
                              hipStream_t stream) {
  const float* x  = (const float*)d_in[0];   // (50000,128)
  const float* W  = (const float*)d_in[1];   // (2,128,64)
  const float* a  = (const float*)d_in[2];   // (2,128,1)
  const int*   ei = (const int*)d_in[3];     // (2,1600000)
  const int* srcIdx = ei;
  const int* dstIdx = ei + N_EDGES;
  float* out = (float*)d_out;                // (50000,128) == Wh layout

  float* sI    = (float*)d_ws;               // N*H floats
  float* sJ    = sI + N_NODES * HEADS;
  float* coeff = sJ + N_NODES * HEADS;
  float* Zs    = coeff + N_NODES * HEADS;    // 2 floats; total ~1.2 MB

  gat_gemm_wmma<<<(N_NODES + 63) / 64, 128, 0, stream>>>(x, W, out);

  const int t2 = N_NODES * HEADS;
  gat_scores<<<(t2 + 255) / 256, 256, 0, stream>>>(out, a, sI, sJ, coeff, Zs);

  gat_edges<<<(N_EDGES + 255) / 256, 256, 0, stream>>>(srcIdx, dstIdx, sI, sJ,
                                                       coeff, Zs);

  gat_scale<<<(N_NODES * (COLS / 4) + 255) / 256, 256, 0, stream>>>(out, coeff, Zs);
}